// LaplacianRegularizer3D_46016279609419
// MI455X (gfx1250) — compile-verified
//
#include <hip/hip_runtime.h>
#include <cstdint>

typedef float v2f __attribute__((ext_vector_type(2)));
typedef float v8f __attribute__((ext_vector_type(8)));

namespace {
constexpr int kBC     = 96;            // B*C = 8*12
constexpr int kD      = 8;
constexpr int kH      = 256;
constexpr int kW      = 256;
constexpr int kTileH  = 32;
constexpr int kLRows  = kTileH + 2;    // 34 rows (1 halo each side)
constexpr int kLPitch = kW + 8;        // 4 zero | 256 data | 4 zero  (16B-aligned float4 reads)
constexpr int kBlocks = kBC * (kH / kTileH);   // 768
}

// Streams each (b,c) volume once from HBM via async DMA into LDS and
// accumulates f*(n3*f - s_cur - 2*s_prev) with a separable 3x3x3 box filter.
__global__ __launch_bounds__(256)
void lap3d_partial(const float* __restrict__ f, float* __restrict__ partials) {
  __shared__ __align__(16) float L[kLRows * kLPitch];
  __shared__ float red[256];

  const int t    = threadIdx.x;
  const int bc   = blockIdx.x / (kH / kTileH);
  const int ht   = blockIdx.x % (kH / kTileH);
  const int h0   = ht * kTileH;
  const int tw   = t & 63;        // 64 column groups of 4
  const int th4  = t >> 6;        // 4 row groups of 8
  const int c0   = tw * 4;
  const int rb   = th4 * 8;       // first LDS row this thread touches

  // One-time LDS zero init: side pads + out-of-range halo rows (never loaded over).
  for (int i = t; i < kLRows * 8; i += 256) {
    const int r = i >> 3, p = i & 7;
    L[r * kLPitch + (p < 4 ? p : 256 + p)] = 0.0f;
  }
  if (h0 == 0)           L[4 + t] = 0.0f;                          // row above h=0
  if (h0 + kTileH == kH) L[(kLRows - 1) * kLPitch + 4 + t] = 0.0f; // row below h=255

  const unsigned lbase = (unsigned)(unsigned long long)(uintptr_t)&L[0];

  float nwj[4];
#pragma unroll
  for (int j = 0; j < 4; ++j) {
    const int cw = c0 + j;
    nwj[j] = (cw == 0 || cw == kW - 1) ? 2.0f : 3.0f;
  }
  float nhi[8];
#pragma unroll
  for (int i = 0; i < 8; ++i) {
    const int ha = h0 + rb + i;
    nhi[i] = (ha == 0 || ha == kH - 1) ? 2.0f : 3.0f;
  }

  float sprev[8][4];
#pragma unroll
  for (int i = 0; i < 8; ++i)
#pragma unroll
    for (int j = 0; j < 4; ++j) sprev[i][j] = 0.0f;

  float acc = 0.0f;

  for (int dp = 0; dp < kD; ++dp) {
    const float* plane = f + ((size_t)bc * kD + dp) * (size_t)(kH * kW);

    // Async DMA: rows h0-1 .. h0+32 of this plane -> LDS (b128 per lane).
#pragma unroll
    for (int k = 0; k < (kLRows * 64 + 255) / 256; ++k) {
      const int i = t + k * 256;
      if (i < kLRows * 64) {
        const int r  = i >> 6;
        const int cg = i & 63;
        const int ha = h0 - 1 + r;
        if (ha >= 0 && ha < kH) {
          const unsigned laddr = lbase + (unsigned)(r * kLPitch + 4 + cg * 4) * 4u;
          const unsigned goff  = (unsigned)(ha * kW + cg * 4) * 4u;
          asm volatile("global_load_async_to_lds_b128 %0, %1, %2"
                       :: "v"(laddr), "v"(goff), "s"(plane)
                       : "memory");
        }
      }
    }
    asm volatile("s_wait_asynccnt 0" ::: "memory");
    __syncthreads();

    const float ndv = (dp == 0 || dp == kD - 1) ? 2.0f : 3.0f;

    float w0[4], w1[4], fprev[4];
#pragma unroll
    for (int r = 0; r < 10; ++r) {
      const float* rowp = &L[(rb + r) * kLPitch + 4 + c0];
      const float4 v  = *(const float4*)rowp;   // 16B aligned ds_load_b128
      const float lft = rowp[-1];
      const float rgt = rowp[4];
      const float s0 = lft + v.x + v.y;
      const float s1 = v.x + v.y + v.z;
      const float s2 = v.y + v.z + v.w;
      const float s3 = v.z + v.w + rgt;

      if (r >= 2) {
        const int i = r - 2;
        const float scur[4] = {w0[0] + w1[0] + s0, w0[1] + w1[1] + s1,
                               w0[2] + w1[2] + s2, w0[3] + w1[3] + s3};
        const float nhn = nhi[i] * ndv;
#pragma unroll
        for (int j = 0; j < 4; ++j) {
          const float fv = fprev[j];                     // f at plane dp, row h0+rb+i
          const float n3 = nhn * nwj[j];
          const float m  = fmaf(-2.0f, sprev[i][j], -scur[j]);
          acc = fmaf(fv, fmaf(n3, fv, m), acc);
          sprev[i][j] = scur[j];
        }
      }
      // rotate row windows
      w0[0] = w1[0]; w0[1] = w1[1]; w0[2] = w1[2]; w0[3] = w1[3];
      w1[0] = s0;    w1[1] = s1;    w1[2] = s2;    w1[3] = s3;
      fprev[0] = v.x; fprev[1] = v.y; fprev[2] = v.z; fprev[3] = v.w;
    }

    __syncthreads();   // LDS reuse next plane
  }

  // Workgroup reduction -> one partial per block.
  red[t] = acc;
  __syncthreads();
#pragma unroll
  for (int s = 128; s > 0; s >>= 1) {
    if (t < s) red[t] += red[t + s];
    __syncthreads();
  }
  if (t == 0) partials[blockIdx.x] = red[0];
}

// Deterministic final reduction; the 256->1 collapse runs on the matrix pipe:
// 4x V_WMMA_F32_16X16X4_F32 with a ones B-matrix accumulates all 256 values
// into every column of the 16x16 accumulator.
__global__ __launch_bounds__(256)
void lap3d_reduce(const float* __restrict__ partials, int n, float* __restrict__ out) {
  __shared__ float sm[256];
  const int t = threadIdx.x;
  float p = 0.0f;
  for (int i = t; i < n; i += 256) p += partials[i];
  sm[t] = p;
  __syncthreads();

  if (t < 32) {                       // wave 0, EXEC all ones
    v8f c = {};
    v2f ones; ones.x = 1.0f; ones.y = 1.0f;
#pragma unroll
    for (int ch = 0; ch < 4; ++ch) {
      v2f a;
      a.x = sm[ch * 64 + t * 2 + 0];
      a.y = sm[ch * 64 + t * 2 + 1];
      c = __builtin_amdgcn_wmma_f32_16x16x4_f32(false, a, false, ones,
                                                (short)0, c, false, false);
    }
    // column n lives in lanes n and n+16; each holds 8 row-sums.
    float s = c[0] + c[1] + c[2] + c[3] + c[4] + c[5] + c[6] + c[7];
    const float tot = s + __shfl_xor(s, 16, 32);
    if (t == 0) out[0] = 2.0f * tot;
  }
}

extern "C" void kernel_launch(void* const* d_in, const int* in_sizes, int n_in,
                              void* d_out, int out_size, void* d_ws, size_t ws_size,
                              hipStream_t stream) {
  const float* f   = (const float*)d_in[0];
  float* out       = (float*)d_out;
  float* partials  = (float*)d_ws;   // kBlocks floats

  lap3d_partial<<<kBlocks, 256, 0, stream>>>(f, partials);
  lap3d_reduce<<<1, 256, 0, stream>>>(partials, kBlocks, out);
}